// MentionScore_72911364817608
// MI455X (gfx1250) — compile-verified
//
#include <hip/hip_runtime.h>
#include <math.h>

typedef float v2f __attribute__((ext_vector_type(2)));
typedef float v8f __attribute__((ext_vector_type(8)));

#define WMMA_F32(a, b, c) \
    __builtin_amdgcn_wmma_f32_16x16x4_f32(false, (a), false, (b), (short)0, (c), false, false)

constexpr int N_T     = 80000;
constexpr int N_S     = 160000;
constexpr int D_STATE = 400;
constexpr int D_EMB   = 350;
constexpr int D_DIST  = 20;
constexpr int H       = 150;   // hidden dim
constexpr int HP      = 160;   // hidden padded to 10 x 16
constexpr int GI      = 1170;  // 2*400 + 350 + 20
constexpr int GI_PAD  = 1172;  // next multiple of 4
constexpr int TILE_M  = 32;    // rows per wave (2 x 16 WMMA A-fragments)
constexpr int WAVES   = 2;     // waves per block
constexpr int NT      = 10;    // n-tiles (150 -> 10 x 16)

// packed-B fragment counts (k-quads)
constexpr int KQ_AW1 = D_STATE / 4;   // 100
constexpr int KQ_H   = HP / 4;        // 40
constexpr int KQ_SW1 = GI_PAD / 4;    // 293

__device__ __forceinline__ v2f ld2(const float* __restrict__ p) {
    return *(const v2f*)p;
}

// ---------------------------------------------------------------------------
// Weight repack: row-major [rows x cols] -> WMMA-B-fragment order.
// out[((kq*10 + nt)*32 + lane)] = { W[kq*4+2*hi][nt*16+lo], W[kq*4+2*hi+1][...] }
// with zero padding baked in. One b64 per lane per fragment in the GEMM loop.
// ---------------------------------------------------------------------------
__global__ __launch_bounds__(256)
void pack_kernel(const float* __restrict__ W, v2f* __restrict__ out,
                 int rows, int cols, int kquads) {
    const int idx   = blockIdx.x * blockDim.x + threadIdx.x;
    const int total = kquads * NT * 32;
    if (idx >= total) return;
    const int lane = idx & 31;
    const int nt   = (idx >> 5) % NT;
    const int kq   = idx / (NT * 32);
    const int lo = lane & 15, hi = lane >> 4;
    const int k   = kq * 4 + 2 * hi;
    const int col = nt * 16 + lo;
    v2f b;
    b[0] = (k     < rows && col < cols) ? W[(k    ) * cols + col] : 0.0f;
    b[1] = (k + 1 < rows && col < cols) ? W[(k + 1) * cols + col] : 0.0f;
    out[idx] = b;
}

// one K-step for a 32-row tile: each B fragment (one b64) feeds TWO WMMAs
__device__ __forceinline__ void gemm_step2(v8f (&acc0)[NT], v8f (&acc1)[NT],
                                           v2f a0, v2f a1,
                                           const v2f* __restrict__ bquad, int lane) {
    const v2f* bb = bquad + lane;
#pragma unroll
    for (int nt = 0; nt < NT; ++nt) {
        v2f b = bb[nt * 32];
        acc0[nt] = WMMA_F32(a0, b, acc0[nt]);
        acc1[nt] = WMMA_F32(a1, b, acc1[nt]);
    }
}

__device__ __forceinline__ void zero_acc(v8f (&acc)[NT]) {
#pragma unroll
    for (int nt = 0; nt < NT; ++nt)
#pragma unroll
        for (int j = 0; j < 8; ++j) acc[nt][j] = 0.0f;
}

// ---------------------------------------------------------------------------
// Kernel 1: alfa[t] = MLP(states[t]) ; one wave handles a 32-token tile.
// ---------------------------------------------------------------------------
__global__ __launch_bounds__(WAVES * 32)
void alfa_kernel(const float* __restrict__ states,
                 const v2f* __restrict__ aW1p, const float* __restrict__ ab1,
                 const v2f* __restrict__ aW2p, const float* __restrict__ ab2,
                 const float* __restrict__ aW3, const float* __restrict__ ab3,
                 float* __restrict__ alfa) {
    __shared__ float lds[WAVES * TILE_M * HP];   // 40 KB: per-wave h1 staging
    const int lane = threadIdx.x & 31;
    const int wave = threadIdx.x >> 5;
    const int lo   = lane & 15;
    const int hi   = lane >> 4;
    float* h = &lds[wave * TILE_M * HP];
    const int rowBase = (blockIdx.x * WAVES + wave) * TILE_M;

    v8f acc0[NT], acc1[NT];
    // ---- layer 1: h1 = relu(states_tile @ aW1 + ab1), K = 400
    zero_acc(acc0);
    zero_acc(acc1);
    const float* arow0 = states + (size_t)(rowBase + lo) * D_STATE;
    const float* arow1 = states + (size_t)(rowBase + 16 + lo) * D_STATE;
    for (int kq = 0; kq < KQ_AW1; ++kq) {
        const int k = kq * 4 + 2 * hi;
        gemm_step2(acc0, acc1, ld2(arow0 + k), ld2(arow1 + k),
                   aW1p + (size_t)kq * (NT * 32), lane);
    }
    // bias + relu -> LDS (C layout -> row-major tile)
#pragma unroll
    for (int nt = 0; nt < NT; ++nt) {
        const int col  = nt * 16 + lo;
        const float b1 = (col < H) ? ab1[col] : 0.0f;
#pragma unroll
        for (int j = 0; j < 8; ++j) {
            float v0 = acc0[nt][j] + b1;
            float v1 = acc1[nt][j] + b1;
            h[(j + 8 * hi) * HP + col]      = v0 > 0.0f ? v0 : 0.0f;
            h[(j + 8 * hi + 16) * HP + col] = v1 > 0.0f ? v1 : 0.0f;
        }
    }

    // ---- layer 2: h2 = h1 @ aW2 + ab2 (relu folded into layer 3), K = 160
    zero_acc(acc0);
    zero_acc(acc1);
    for (int kq = 0; kq < KQ_H; ++kq) {
        const int k = kq * 4 + 2 * hi;
        gemm_step2(acc0, acc1, ld2(&h[lo * HP + k]), ld2(&h[(lo + 16) * HP + k]),
                   aW2p + (size_t)kq * (NT * 32), lane);
    }

    // ---- layer 3: alfa = relu(h2 + ab2) @ aW3 + ab3 (register dot + shuffle tree)
    float part0[8], part1[8];
#pragma unroll
    for (int j = 0; j < 8; ++j) { part0[j] = 0.0f; part1[j] = 0.0f; }
#pragma unroll
    for (int nt = 0; nt < NT; ++nt) {
        const int col  = nt * 16 + lo;
        const float b2 = (col < H) ? ab2[col] : 0.0f;
        const float w3 = (col < H) ? aW3[col] : 0.0f;
#pragma unroll
        for (int j = 0; j < 8; ++j) {
            float v0 = acc0[nt][j] + b2;
            float v1 = acc1[nt][j] + b2;
            v0 = v0 > 0.0f ? v0 : 0.0f;
            v1 = v1 > 0.0f ? v1 : 0.0f;
            part0[j] += v0 * w3;
            part1[j] += v1 * w3;
        }
    }
#pragma unroll
    for (int j = 0; j < 8; ++j) {
#pragma unroll
        for (int m = 1; m <= 8; m <<= 1) {
            part0[j] += __shfl_xor(part0[j], m, 32);
            part1[j] += __shfl_xor(part1[j], m, 32);
        }
    }
    if (lo == 0) {
        const float b3 = ab3[0];
#pragma unroll
        for (int j = 0; j < 8; ++j) {
            alfa[rowBase + 8 * hi + j]      = part0[j] + b3;
            alfa[rowBase + 16 + 8 * hi + j] = part1[j] + b3;
        }
    }
}

// ---------------------------------------------------------------------------
// Kernel 2: build g_i on the fly as WMMA A-fragments (write to d_out once),
// run the mention MLP, write scores. One wave per 32-span tile.
// K-loop is split into uniform segments (no per-iteration divergence).
// ---------------------------------------------------------------------------
__global__ __launch_bounds__(WAVES * 32)
void span_kernel(const float* __restrict__ states, const float* __restrict__ embeds,
                 const int* __restrict__ span_starts, const int* __restrict__ span_widths,
                 const float* __restrict__ width_table, const float* __restrict__ alfa,
                 const v2f* __restrict__ sW1p, const float* __restrict__ sb1,
                 const v2f* __restrict__ sW2p, const float* __restrict__ sb2,
                 const float* __restrict__ sW3, const float* __restrict__ sb3,
                 float* __restrict__ gi_out, float* __restrict__ score_out) {
    __shared__ float lds[WAVES * TILE_M * HP];   // 40 KB: per-wave h1 staging
    const int lane = threadIdx.x & 31;
    const int wave = threadIdx.x >> 5;
    const int lo   = lane & 15;
    const int hi   = lane >> 4;
    float* h = &lds[wave * TILE_M * HP];
    const int spanBase = (blockIdx.x * WAVES + wave) * TILE_M;

    // per-span scalars for the lane's two spans (lanes l and l+16 match)
    const int sp0 = spanBase + lo;
    const int sp1 = spanBase + 16 + lo;
    const int s0   = span_starts[sp0];
    const int s1   = span_starts[sp1];
    const int wdt0 = span_widths[sp0];          // 0 or 1
    const int wdt1 = span_widths[sp1];
    const float ea0 = expf(alfa[s0]);
    const float eb0 = wdt0 ? expf(alfa[s0 + 1]) : 0.0f;
    const float ea1 = expf(alfa[s1]);
    const float eb1 = wdt1 ? expf(alfa[s1 + 1]) : 0.0f;
    const float w00 = ea0 / (ea0 + eb0);        // exact 1.0 when wdt == 0
    const float w01 = eb0 / (ea0 + eb0);
    const float w10 = ea1 / (ea1 + eb1);
    const float w11 = eb1 / (ea1 + eb1);

    const float* srow0 = states + (size_t)s0 * D_STATE;
    const float* srow1 = states + (size_t)s1 * D_STATE;
    const float* erow0 = states + (size_t)(s0 + wdt0) * D_STATE;
    const float* erow1 = states + (size_t)(s1 + wdt1) * D_STATE;
    const float* m00   = embeds + (size_t)s0 * D_EMB;
    const float* m01   = embeds + (size_t)(s0 + 1) * D_EMB;
    const float* m10   = embeds + (size_t)s1 * D_EMB;
    const float* m11   = embeds + (size_t)(s1 + 1) * D_EMB;
    const float* wrow0 = width_table + (wdt0 + 1) * D_DIST;   // stoi(len): 1->1, 2->2
    const float* wrow1 = width_table + (wdt1 + 1) * D_DIST;
    float* girow0 = gi_out + (size_t)sp0 * GI;
    float* girow1 = gi_out + (size_t)sp1 * GI;

    v8f acc0[NT], acc1[NT];
    // ---- layer 1: h1 = relu(g_i @ sW1 + sb1), K = 1170 (pad 1172)
    zero_acc(acc0);
    zero_acc(acc1);
    int kq = 0;
    // segment 1: states[start], k in [0,400)
    for (int k0 = 0; k0 < 400; k0 += 4, ++kq) {
        const int k = k0 + 2 * hi;
        v2f a0 = ld2(srow0 + k);
        v2f a1 = ld2(srow1 + k);
        *(v2f*)(girow0 + k) = a0;
        *(v2f*)(girow1 + k) = a1;
        gemm_step2(acc0, acc1, a0, a1, sW1p + (size_t)kq * (NT * 32), lane);
    }
    // segment 2: states[end], k in [400,800)
    for (int k0 = 400; k0 < 800; k0 += 4, ++kq) {
        const int k = k0 + 2 * hi;
        v2f a0 = ld2(erow0 + (k - 400));
        v2f a1 = ld2(erow1 + (k - 400));
        *(v2f*)(girow0 + k) = a0;
        *(v2f*)(girow1 + k) = a1;
        gemm_step2(acc0, acc1, a0, a1, sW1p + (size_t)kq * (NT * 32), lane);
    }
    // segment 3: weighted embeds, k in [800,1148)
    for (int k0 = 800; k0 < 1148; k0 += 4, ++kq) {
        const int k = k0 + 2 * hi;
        v2f x00 = ld2(m00 + (k - 800));
        v2f x01 = ld2(m01 + (k - 800));
        v2f x10 = ld2(m10 + (k - 800));
        v2f x11 = ld2(m11 + (k - 800));
        v2f a0, a1;
        a0[0] = w00 * x00[0] + w01 * x01[0];
        a0[1] = w00 * x00[1] + w01 * x01[1];
        a1[0] = w10 * x10[0] + w11 * x11[0];
        a1[1] = w10 * x10[1] + w11 * x11[1];
        *(v2f*)(girow0 + k) = a0;
        *(v2f*)(girow1 + k) = a1;
        gemm_step2(acc0, acc1, a0, a1, sW1p + (size_t)kq * (NT * 32), lane);
    }
    // boundary k0 = 1148: hi=0 -> embeds[348,349], hi=1 -> width[0,1]
    {
        const int k = 1148 + 2 * hi;
        v2f a0, a1;
        if (hi == 0) {
            v2f x00 = ld2(m00 + 348);
            v2f x01 = ld2(m01 + 348);
            v2f x10 = ld2(m10 + 348);
            v2f x11 = ld2(m11 + 348);
            a0[0] = w00 * x00[0] + w01 * x01[0];
            a0[1] = w00 * x00[1] + w01 * x01[1];
            a1[0] = w10 * x10[0] + w11 * x11[0];
            a1[1] = w10 * x10[1] + w11 * x11[1];
        } else {
            a0 = ld2(wrow0 + 0);
            a1 = ld2(wrow1 + 0);
        }
        *(v2f*)(girow0 + k) = a0;
        *(v2f*)(girow1 + k) = a1;
        gemm_step2(acc0, acc1, a0, a1, sW1p + (size_t)kq * (NT * 32), lane);
        ++kq;
    }
    // segment 4: width embedding, k in [1152,1168)
    for (int k0 = 1152; k0 < 1168; k0 += 4, ++kq) {
        const int k = k0 + 2 * hi;
        v2f a0 = ld2(wrow0 + (k - 1150));
        v2f a1 = ld2(wrow1 + (k - 1150));
        *(v2f*)(girow0 + k) = a0;
        *(v2f*)(girow1 + k) = a1;
        gemm_step2(acc0, acc1, a0, a1, sW1p + (size_t)kq * (NT * 32), lane);
    }
    // boundary k0 = 1168: hi=0 -> width[18,19], hi=1 -> zero pad (no store)
    {
        v2f a0, a1;
        if (hi == 0) {
            a0 = ld2(wrow0 + 18);
            a1 = ld2(wrow1 + 18);
            *(v2f*)(girow0 + 1168) = a0;
            *(v2f*)(girow1 + 1168) = a1;
        } else {
            a0[0] = 0.0f; a0[1] = 0.0f;
            a1[0] = 0.0f; a1[1] = 0.0f;
        }
        gemm_step2(acc0, acc1, a0, a1, sW1p + (size_t)kq * (NT * 32), lane);
        ++kq;
    }

#pragma unroll
    for (int nt = 0; nt < NT; ++nt) {
        const int col  = nt * 16 + lo;
        const float b1 = (col < H) ? sb1[col] : 0.0f;
#pragma unroll
        for (int j = 0; j < 8; ++j) {
            float v0 = acc0[nt][j] + b1;
            float v1 = acc1[nt][j] + b1;
            h[(j + 8 * hi) * HP + col]      = v0 > 0.0f ? v0 : 0.0f;
            h[(j + 8 * hi + 16) * HP + col] = v1 > 0.0f ? v1 : 0.0f;
        }
    }

    // ---- layer 2: K = 160
    zero_acc(acc0);
    zero_acc(acc1);
    for (int q = 0; q < KQ_H; ++q) {
        const int k = q * 4 + 2 * hi;
        gemm_step2(acc0, acc1, ld2(&h[lo * HP + k]), ld2(&h[(lo + 16) * HP + k]),
                   sW2p + (size_t)q * (NT * 32), lane);
    }

    // ---- layer 3: scores = relu(h2 + sb2) @ sW3 + sb3
    float part0[8], part1[8];
#pragma unroll
    for (int j = 0; j < 8; ++j) { part0[j] = 0.0f; part1[j] = 0.0f; }
#pragma unroll
    for (int nt = 0; nt < NT; ++nt) {
        const int col  = nt * 16 + lo;
        const float b2 = (col < H) ? sb2[col] : 0.0f;
        const float w3 = (col < H) ? sW3[col] : 0.0f;
#pragma unroll
        for (int j = 0; j < 8; ++j) {
            float v0 = acc0[nt][j] + b2;
            float v1 = acc1[nt][j] + b2;
            v0 = v0 > 0.0f ? v0 : 0.0f;
            v1 = v1 > 0.0f ? v1 : 0.0f;
            part0[j] += v0 * w3;
            part1[j] += v1 * w3;
        }
    }
#pragma unroll
    for (int j = 0; j < 8; ++j) {
#pragma unroll
        for (int m = 1; m <= 8; m <<= 1) {
            part0[j] += __shfl_xor(part0[j], m, 32);
            part1[j] += __shfl_xor(part1[j], m, 32);
        }
    }
    if (lo == 0) {
        const float b3 = sb3[0];
#pragma unroll
        for (int j = 0; j < 8; ++j) {
            score_out[spanBase + 8 * hi + j]      = part0[j] + b3;
            score_out[spanBase + 16 + 8 * hi + j] = part1[j] + b3;
        }
    }
}

extern "C" void kernel_launch(void* const* d_in, const int* in_sizes, int n_in,
                              void* d_out, int out_size, void* d_ws, size_t ws_size,
                              hipStream_t stream) {
    const float* states      = (const float*)d_in[0];
    const float* embeds      = (const float*)d_in[1];
    const int*   span_starts = (const int*)d_in[2];
    const int*   span_widths = (const int*)d_in[3];
    const float* aW1 = (const float*)d_in[4];
    const float* ab1 = (const float*)d_in[5];
    const float* aW2 = (const float*)d_in[6];
    const float* ab2 = (const float*)d_in[7];
    const float* aW3 = (const float*)d_in[8];
    const float* ab3 = (const float*)d_in[9];
    const float* wtab = (const float*)d_in[10];
    const float* sW1 = (const float*)d_in[11];
    const float* sb1 = (const float*)d_in[12];
    const float* sW2 = (const float*)d_in[13];
    const float* sb2 = (const float*)d_in[14];
    const float* sW3 = (const float*)d_in[15];
    const float* sb3 = (const float*)d_in[16];

    // workspace layout (all offsets 8-byte aligned)
    char* ws = (char*)d_ws;
    float* alfa = (float*)ws;                               // 80000 f32 = 320000 B
    v2f* aW1p = (v2f*)(ws + 320000);                        // 100*320 fragments
    v2f* aW2p = (v2f*)(ws + 320000 + 256000);               //  40*320
    v2f* sW1p = (v2f*)(ws + 320000 + 256000 + 102400);      // 293*320
    v2f* sW2p = (v2f*)(ws + 320000 + 256000 + 102400 + 750080);   // 40*320

    // repack weights into B-fragment order (cheap; runs every call, deterministic)
    auto packGrid = [](int kq) { return (kq * NT * 32 + 255) / 256; };
    pack_kernel<<<packGrid(KQ_AW1), 256, 0, stream>>>(aW1, aW1p, D_STATE, H, KQ_AW1);
    pack_kernel<<<packGrid(KQ_H),   256, 0, stream>>>(aW2, aW2p, H,       H, KQ_H);
    pack_kernel<<<packGrid(KQ_SW1), 256, 0, stream>>>(sW1, sW1p, GI,      H, KQ_SW1);
    pack_kernel<<<packGrid(KQ_H),   256, 0, stream>>>(sW2, sW2p, H,       H, KQ_H);

    float* gi     = (float*)d_out;                // [160000 x 1170]
    float* scores = gi + (size_t)N_S * GI;        // [160000]

    alfa_kernel<<<N_T / (WAVES * TILE_M), WAVES * 32, 0, stream>>>(
        states, aW1p, ab1, aW2p, ab2, aW3, ab3, alfa);

    span_kernel<<<N_S / (WAVES * TILE_M), WAVES * 32, 0, stream>>>(
        states, embeds, span_starts, span_widths, wtab, alfa,
        sW1p, sb1, sW2p, sb2, sW3, sb3, gi, scores);
}